// ColorReducer_32289564131650
// MI455X (gfx1250) — compile-verified
//
#include <hip/hip_runtime.h>
#include <hip/hip_bf16.h>

typedef __attribute__((ext_vector_type(2))) float v2f;
typedef __attribute__((ext_vector_type(8))) float v8f;

// Nearest-palette-color reducer using V_WMMA_F32_16X16X4_F32.
//
//   A (16x4, M=palette color k, K=channel): [pR, pG, pB, -0.5*|p|^2]   (loop-invariant)
//   B (4x16, K=channel, N=pixel):           [xR, xG, xB, 1.0]
//   D[k][n] = x_n . p_k - 0.5*|p_k|^2  ->  argmax_k D == argmin_k ||x - p_k||^2
//
// C/D layout (32-bit 16x16): lane n<16 holds colors 0-7 for pixel n in its 8
// VGPRs; lane n+16 holds colors 8-15 for the same pixel. In-lane argmax +
// one shfl_xor(16) merge finishes the argmin. Two WMMAs cover 32 pixels and
// leave lane l owning pixel base+l -> fully coalesced stores.
//
// Each wave processes 8 chunks of 32 pixels (256 px) so the prologue
// (palette LDS fill, A-matrix build, address math) is amortized and chunk
// offsets fold into vmem immediate offsets.
__global__ void ColorReducer_32289564131650_kernel(
    const float* __restrict__ x,    // (B,3,H,W) planar
    const float* __restrict__ pal,  // (16,3)
    float* __restrict__ out,        // (B,3,H,W) planar
    int HW)                         // H*W (multiple of 256)
{
    __shared__ float lpal[16 * 4];  // 16B stride -> ds_load_b128 lookup

    const int t = threadIdx.x;
    if (t < 16) {
        lpal[t * 4 + 0] = pal[t * 3 + 0];
        lpal[t * 4 + 1] = pal[t * 3 + 1];
        lpal[t * 4 + 2] = pal[t * 3 + 2];
        lpal[t * 4 + 3] = 0.0f;
    }
    __syncthreads();

    const int lane = t & 31;
    const int wave = t >> 5;
    const int li   = lane & 15;
    const bool lo  = (lane < 16);

    // ---- A matrix: palette (16 colors x K=4), built once ----
    // lane<16 : VGPR0 = K0 = p[li].r , VGPR1 = K1 = p[li].g
    // lane>=16: VGPR0 = K2 = p[li].b , VGPR1 = K3 = -0.5*|p[li]|^2
    const float p0 = lpal[li * 4 + 0];
    const float p1 = lpal[li * 4 + 1];
    const float p2 = lpal[li * 4 + 2];
    const float pn = -0.5f * (p0 * p0 + p1 * p1 + p2 * p2);
    v2f A;
    A.x = lo ? p0 : p2;
    A.y = lo ? p1 : pn;

    // ---- per-wave pixel range: 256 consecutive pixels, one batch ----
    const long gp0   = ((long)blockIdx.x * (blockDim.x >> 5) + wave) * 256L;
    const long batch = gp0 / HW;
    const long n0    = gp0 - batch * (long)HW;
    const float* xb  = x + batch * 3L * HW + n0;
    float* ob        = out + batch * 3L * HW + n0 + lane;

    const long offc = lo ? 0L : 2L * HW;  // R-plane vs B-plane
    const int  kb   = lo ? 0 : 8;
    const v8f  cz   = {};

#pragma unroll
    for (int u = 0; u < 8; ++u) {
        const int o = u * 32;  // folds into vmem immediate offsets

        // B: lane<16: {K0=R[n], K1=G[n]}  lane>=16: {K2=B[n], K3=1.0}
        v2f B1, B2;
        B1.x = xb[offc + li + o];
        const float g1 = xb[(long)HW + li + o];       // unconditional, masked by select
        B1.y = lo ? g1 : 1.0f;
        B2.x = xb[offc + 16 + li + o];
        const float g2 = xb[(long)HW + 16 + li + o];
        B2.y = lo ? g2 : 1.0f;

        // 8 args: (neg_a, A, neg_b, B, c_mod, C, reuse_a, reuse_b)
        v8f d1 = __builtin_amdgcn_wmma_f32_16x16x4_f32(false, A, false, B1, (short)0, cz, false, false);
        v8f d2 = __builtin_amdgcn_wmma_f32_16x16x4_f32(false, A, false, B2, (short)0, cz, false, false);

        // in-lane argmax over 8 colors (strict > keeps lowest k on ties)
        float best1 = d1[0];
        int   idx1  = kb;
#pragma unroll
        for (int r = 1; r < 8; ++r) {
            const bool c = d1[r] > best1;
            best1 = c ? d1[r] : best1;
            idx1  = c ? (kb + r) : idx1;
        }
        float best2 = d2[0];
        int   idx2  = kb;
#pragma unroll
        for (int r = 1; r < 8; ++r) {
            const bool c = d2[r] > best2;
            best2 = c ? d2[r] : best2;
            idx2  = c ? (kb + r) : idx2;
        }

        // merge colors 0-7 (lanes 0-15) with colors 8-15 (lanes 16-31)
        {
            const float ov = __shfl_xor(best1, 16, 32);
            const int   oi = __shfl_xor(idx1, 16, 32);
            const bool  tk = (ov > best1) || ((ov == best1) && (oi < idx1));
            best1 = tk ? ov : best1;
            idx1  = tk ? oi : idx1;
        }
        {
            const float ov = __shfl_xor(best2, 16, 32);
            const int   oi = __shfl_xor(idx2, 16, 32);
            const bool  tk = (ov > best2) || ((ov == best2) && (oi < idx2));
            best2 = tk ? ov : best2;
            idx2  = tk ? oi : idx2;
        }

        // lane l<16 owns pixel n0+o+l (WMMA1); lane l>=16 owns pixel n0+o+l (WMMA2)
        const int fidx = lo ? idx1 : idx2;
        const float4 col = *(const float4*)(lpal + fidx * 4);   // ds_load_b128

        ob[o]            = col.x;
        ob[(long)HW + o] = col.y;
        ob[2L * HW + o]  = col.z;
    }
}

extern "C" void kernel_launch(void* const* d_in, const int* in_sizes, int n_in,
                              void* d_out, int out_size, void* d_ws, size_t ws_size,
                              hipStream_t stream) {
    const float* x   = (const float*)d_in[0];   // (16,3,512,512) f32
    const float* pal = (const float*)d_in[1];   // (16,3) f32
    float* out       = (float*)d_out;           // (16,3,512,512) f32

    const int  HW      = 512 * 512;
    const long totalP  = (long)in_sizes[0] / 3;     // 4,194,304 pixels
    const int  threads = 256;                       // 8 waves
    const long pixPerBlock = 8L * 256L;             // 8 waves x 256 px/wave
    const int  blocks  = (int)(totalP / pixPerBlock);

    ColorReducer_32289564131650_kernel<<<blocks, threads, 0, stream>>>(x, pal, out, HW);
}